// GNN_49280454754828
// MI455X (gfx1250) — compile-verified
//
#include <hip/hip_runtime.h>

#define NN 100000
#define NE 1250000
#define HID 64
#define TSTRIDE 68   // row stride in LDS: (m*68+k)%64 unique per lane -> no bank conflicts

typedef __attribute__((ext_vector_type(2))) float v2f;
typedef __attribute__((ext_vector_type(8))) float v8f;

__global__ __launch_bounds__(256) void gnn_zero_kernel(float* __restrict__ p, long long n) {
  long long i = (long long)blockIdx.x * blockDim.x + threadIdx.x;
  const long long stride = (long long)gridDim.x * blockDim.x;
  for (; i < n; i += stride) p[i] = 0.0f;
}

// 16 threads per edge; each loads a float4 of the source row and atomically
// accumulates into the destination row. Lane q==0 counts the edge.
__global__ __launch_bounds__(256) void gnn_scatter_kernel(
    const float* __restrict__ x, const int* __restrict__ ei,
    float* __restrict__ agg, float* __restrict__ cnt) {
  const long long idx = (long long)blockIdx.x * blockDim.x + threadIdx.x;
  const int e = (int)(idx >> 4);
  const int q = (int)(idx & 15);
  if (e >= NE) return;
  const int s = ei[e];
  const int d = ei[NE + e];
  const float4 v = *(const float4*)(x + (size_t)s * HID + q * 4);
  float* a = agg + (size_t)d * HID + q * 4;
  atomicAdd(a + 0, v.x);
  atomicAdd(a + 1, v.y);
  atomicAdd(a + 2, v.z);
  atomicAdd(a + 3, v.w);
  if (q == 0) atomicAdd(cnt + d, 1.0f);
}

// One wave per 16-node tile: out[16x64] = mean[16x64]@Wl + bl + xr[16x64]@Wr,
// computed with full-fp32 WMMA (v_wmma_f32_16x16x4_f32), optional fused ReLU.
__global__ __launch_bounds__(256) void gnn_sage_wmma_kernel(
    const float* __restrict__ agg, const float* __restrict__ cnt,
    const float* __restrict__ xr,
    const float* __restrict__ Wl, const float* __restrict__ bl,
    const float* __restrict__ Wr,
    float* __restrict__ out, int do_relu) {
  __shared__ float sM[8 * 16 * TSTRIDE];  // mean tiles, one per wave
  __shared__ float sX[8 * 16 * TSTRIDE];  // root tiles
  const int wave = threadIdx.x >> 5;
  const int lane = threadIdx.x & 31;
  const int tile0 = (blockIdx.x * 8 + wave) * 16;
  float* wM = sM + wave * 16 * TSTRIDE;
  float* wX = sX + wave * 16 * TSTRIDE;

  // Stage 16x64 mean and root tiles into LDS (2 lanes per row, float4 chunks).
  {
    const int r = lane >> 1;
    const int c0 = (lane & 1) * 32;
    int node = tile0 + r;
    if (node >= NN) node = NN - 1;  // clamp; garbage rows never stored
    const float inv = 1.0f / fmaxf(cnt[node], 1.0f);
    const float4* ap = (const float4*)(agg + (size_t)node * HID + c0);
    const float4* xp = (const float4*)(xr + (size_t)node * HID + c0);
    float4* dm = (float4*)(wM + r * TSTRIDE + c0);
    float4* dx = (float4*)(wX + r * TSTRIDE + c0);
#pragma unroll
    for (int i = 0; i < 8; ++i) {
      float4 a = ap[i];
      a.x *= inv; a.y *= inv; a.z *= inv; a.w *= inv;
      dm[i] = a;
      dx[i] = xp[i];
    }
  }
  __syncthreads();

  const int mc = lane & 15;   // A-row index == C/D column index
  const int kh = lane >> 4;   // lane half selects K offset (A/B) and M offset (C/D)

  // Accumulators for the 4 column tiles, initialized with the lin_l bias.
  v8f acc[4];
#pragma unroll
  for (int t = 0; t < 4; ++t) {
    const float b = bl[t * 16 + mc];
#pragma unroll
    for (int r = 0; r < 8; ++r) acc[t][r] = b;
  }

#pragma unroll
  for (int k0 = 0; k0 < HID; k0 += 4) {
    const int kc = k0 + 2 * kh;
    v2f aM, aX;
    aM[0] = wM[mc * TSTRIDE + kc];
    aM[1] = wM[mc * TSTRIDE + kc + 1];
    aX[0] = wX[mc * TSTRIDE + kc];
    aX[1] = wX[mc * TSTRIDE + kc + 1];
#pragma unroll
    for (int t = 0; t < 4; ++t) {
      const int col = t * 16 + mc;
      v2f bL, bR;
      bL[0] = Wl[(kc + 0) * HID + col];
      bL[1] = Wl[(kc + 1) * HID + col];
      bR[0] = Wr[(kc + 0) * HID + col];
      bR[1] = Wr[(kc + 1) * HID + col];
      acc[t] = __builtin_amdgcn_wmma_f32_16x16x4_f32(
          false, aM, false, bL, (short)0, acc[t], false, false);
      acc[t] = __builtin_amdgcn_wmma_f32_16x16x4_f32(
          false, aX, false, bR, (short)0, acc[t], false, false);
    }
  }

  // C/D layout: VGPR r holds M=r (lanes 0-15) / M=r+8 (lanes 16-31), N=lane&15.
#pragma unroll
  for (int t = 0; t < 4; ++t) {
#pragma unroll
    for (int r = 0; r < 8; ++r) {
      const int node = tile0 + r + 8 * kh;
      if (node < NN) {
        float v = acc[t][r];
        if (do_relu) v = fmaxf(v, 0.0f);
        out[(size_t)node * HID + t * 16 + mc] = v;
      }
    }
  }
}

extern "C" void kernel_launch(void* const* d_in, const int* in_sizes, int n_in,
                              void* d_out, int out_size, void* d_ws, size_t ws_size,
                              hipStream_t stream) {
  const float* x   = (const float*)d_in[0];
  const int*   ei  = (const int*)d_in[1];   // edge_index [2, NE], int32
  const float* W1l = (const float*)d_in[2];
  const float* b1l = (const float*)d_in[3];
  const float* W1r = (const float*)d_in[4];
  const float* W2l = (const float*)d_in[5];
  const float* b2l = (const float*)d_in[6];
  const float* W2r = (const float*)d_in[7];
  float* out = (float*)d_out;

  float* agg = (float*)d_ws;                       // NN*HID floats
  float* cnt = agg + (size_t)NN * HID;             // NN floats

  const long long zn = (long long)NN * HID + NN;
  const int scatter_blocks = (NE * 16) / 256;      // 78125 exactly
  const int sage_blocks = (NN + 127) / 128;        // 8 tiles of 16 nodes per block

  // ---- Layer 1: h = relu(mean@W1l + b1l + x@W1r), h stored in d_out ----
  gnn_zero_kernel<<<2048, 256, 0, stream>>>(agg, zn);
  gnn_scatter_kernel<<<scatter_blocks, 256, 0, stream>>>(x, ei, agg, cnt);
  gnn_sage_wmma_kernel<<<sage_blocks, 256, 0, stream>>>(agg, cnt, x, W1l, b1l, W1r,
                                                        out, 1);
  // ---- Layer 2: out = mean_h@W2l + b2l + h@W2r (in-place over h) ----
  gnn_zero_kernel<<<2048, 256, 0, stream>>>(agg, zn);
  gnn_scatter_kernel<<<scatter_blocks, 256, 0, stream>>>(out, ei, agg, cnt);
  gnn_sage_wmma_kernel<<<sage_blocks, 256, 0, stream>>>(agg, cnt, out, W2l, b2l, W2r,
                                                        out, 0);
}